// glimpse_network_54726473285977
// MI455X (gfx1250) — compile-verified
//
#include <hip/hip_runtime.h>

// ---------------------------------------------------------------------------
// Glimpse network for MI455X (gfx1250, wave32).
//   phi = multiscale glimpse (B=128, D=36864)       -> d_ws (18.9 MB)
//   out[:, :128] = relu(phi @ W1 + b1)              -> V_WMMA_F32_16X16X4_F32,
//                  split-K x8, LDS double-buffered via GLOBAL_LOAD_ASYNC_TO_LDS
//   out[:, 128:] = relu(l @ W2 + b2)                -> trivial VALU
// Roofline: ~170 MB total HBM+L2 traffic @ 23.3 TB/s => memory bound; exact
// f32 WMMA keeps full precision while using the matrix pipe.
// ---------------------------------------------------------------------------

typedef __attribute__((ext_vector_type(2))) float v2f;
typedef __attribute__((ext_vector_type(8))) float v8f;

#define B_SZ  128
#define H_SZ  256
#define W_SZ  256
#define C_SZ  3
#define G_SZ  64
#define K_SC  3
#define D_SZ  (K_SC * G_SZ * G_SZ * C_SZ)   // 36864
#define HG    128
#define HL    128
#define OUTW  (HG + HL)                     // 256

#define KSPLIT 8                 // split-K factor -> 8x8 = 64 blocks
#define DSPL   (D_SZ / KSPLIT)   // 4608 K per split
#define KCH    128               // K chunk staged in LDS
#define NCH    (DSPL / KCH)      // 36 chunks per split
#define ASTR   132               // A row stride (dwords): 16B-aligned, banks 4*lm+kb all distinct
#define BSTR   136               // B row stride (dwords): 2*136%64=16 -> lane halves on disjoint banks

// --------- CDNA5 async memory->LDS path (guarded; falls back cleanly) ------
#if defined(__has_builtin)
#if __has_builtin(__builtin_amdgcn_global_load_async_to_lds_b128)
#define HAVE_ASYNC 1
#endif
#endif
#ifndef HAVE_ASYNC
#define HAVE_ASYNC 0
#endif

#if HAVE_ASYNC
// Builtin signature (from hipcc diagnostic): param 0 is
// 'int __vector(4) __device__ *'  -> 16B int vector in addrspace(1);
// param 1 is the LDS-side pointer (addrspace(3)).
typedef __attribute__((__vector_size__(4 * sizeof(int)))) int i128_t;
typedef __attribute__((address_space(1))) i128_t i128_g;
typedef __attribute__((address_space(3))) i128_t i128_l;
#endif

__device__ __forceinline__ void cp16_to_lds(float* lds, const float* g) {
#if HAVE_ASYNC
  // GLOBAL_LOAD_ASYNC_TO_LDS_B128: no VGPR staging, tracked by ASYNCcnt.
  // Generic->AS casts via integer: AS1 bits == flat bits; AS3 = low 32 bits.
  __builtin_amdgcn_global_load_async_to_lds_b128(
      (i128_g*)(unsigned long long)g,
      (i128_l*)(unsigned int)(unsigned long long)lds,
      0, 0);
#else
  *(float4*)lds = *(const float4*)g;
#endif
}

__device__ __forceinline__ void wait_async0() {
#if HAVE_ASYNC
#if __has_builtin(__builtin_amdgcn_s_wait_asynccnt)
  __builtin_amdgcn_s_wait_asynccnt(0);
#else
  asm volatile("s_wait_asynccnt 0x0" ::: "memory");
#endif
#endif
}

// ---------------- Kernel 1: glimpse extraction (memory bound) --------------
// One block per (batch, scale). Matches jax.lax.dynamic_slice clamping and
// the (G, f, G, f, C).mean(axis=(2,4)) pooling exactly.
__global__ void __launch_bounds__(256)
glimpse_extract(const float* __restrict__ x,
                const float* __restrict__ l,
                float* __restrict__ phi) {
  const int blk = blockIdx.x;          // b * K_SC + scale
  const int b = blk / K_SC;
  const int i = blk % K_SC;
  const int f = 1 << i;                // pooling factor (1, 2, 4)
  const int size = G_SZ * f;           // patch size (64, 128, 256)

  const float lx = l[b * 2 + 0];
  const float ly = l[b * 2 + 1];
  // astype(int32) on a non-negative float truncates toward zero == C cast.
  const int cx = (int)(lx * (float)(H_SZ / 2) + (float)(H_SZ / 2));
  const int cy = (int)(ly * (float)(W_SZ / 2) + (float)(W_SZ / 2));
  int px = cx - size / 2;  px = min(max(px, 0), H_SZ - size);  // dynamic_slice clamp
  int py = cy - size / 2;  py = min(max(py, 0), W_SZ - size);

  const float inv = 1.0f / (float)(f * f);
  const float* __restrict__ xb = x + (size_t)b * H_SZ * W_SZ * C_SZ;
  float* __restrict__ pb = phi + (size_t)b * D_SZ + (size_t)i * G_SZ * G_SZ * C_SZ;

  for (int t = threadIdx.x; t < G_SZ * G_SZ * C_SZ; t += 256) {
    const int c  = t % C_SZ;
    const int g2 = (t / C_SZ) % G_SZ;
    const int g1 = t / (C_SZ * G_SZ);
    const int r0 = px + g1 * f;
    const int c0 = py + g2 * f;
    float s = 0.0f;
    for (int u = 0; u < f; ++u) {
      const float* __restrict__ row = xb + ((size_t)(r0 + u) * W_SZ + c0) * C_SZ + c;
      for (int v = 0; v < f; ++v) s += row[v * C_SZ];
    }
    pb[t] = s * inv;
  }
}

// ---------------- Kernel 2: split-K WMMA GEMM partials ---------------------
// Grid (8 M-tiles, 8 K-splits), block = 256 threads = 8 waves; wave w owns
// N tile [16w, 16w+16). Both A (phi) and B (W1) chunks are staged in LDS via
// coalesced b128 async copies, double buffered; in-loop operands come from
// conflict-free DS reads. Per ISA 7.12.2 (32-bit layouts): lane {hi=lane>>4,
// lm=lane&15} holds K pair kb=kk+2*hi, kb+1 for both A (row lm) and B (col lm).
__global__ void __launch_bounds__(256)
glimpse_fc_wmma(const float* __restrict__ phi,
                const float* __restrict__ W1,
                float* __restrict__ part) {
  __shared__ float As[2][16 * ASTR];    //  2 x  8.4 KB
  __shared__ float Bs[2][KCH * BSTR];   //  2 x 68.0 KB   (total ~153 KB < 320 KB)

  const int m0    = blockIdx.x * 16;
  const int split = blockIdx.y;
  const int k0    = split * DSPL;
  const int tid   = threadIdx.x;
  const int wave  = tid >> 5;
  const int lane  = tid & 31;
  const int hi    = lane >> 4;
  const int lm    = lane & 15;
  const int n0    = wave * 16;

  auto stage = [&](int buf, int kc) {
    // A tile: 16 rows x KCH cols = 512 float4 -> 2 per thread, coalesced.
#pragma unroll
    for (int j = 0; j < 2; ++j) {
      const int e = tid + 256 * j, r = e >> 5, q = (e & 31) << 2;
      cp16_to_lds(&As[buf][r * ASTR + q], &phi[(size_t)(m0 + r) * D_SZ + kc + q]);
    }
    // B tile: KCH rows x 128 cols = 4096 float4 -> 16 per thread, coalesced.
#pragma unroll
    for (int j = 0; j < 16; ++j) {
      const int e = tid + 256 * j, r = e >> 5, q = (e & 31) << 2;
      cp16_to_lds(&Bs[buf][r * BSTR + q], &W1[(size_t)(kc + r) * HG + q]);
    }
  };

  v8f acc = {};  // C/D: 16x16 f32 in 8 VGPRs

  stage(0, k0);
  wait_async0();
  __syncthreads();

  for (int c = 0; c < NCH; ++c) {
    const int cur = c & 1;
    if (c + 1 < NCH) stage(cur ^ 1, k0 + (c + 1) * KCH);  // overlap with compute

    const float* __restrict__ Ab = &As[cur][lm * ASTR];
    const float* __restrict__ Bb = &Bs[cur][n0 + lm];
#pragma unroll 8
    for (int kk = 0; kk < KCH; kk += 4) {
      const int kb = kk + 2 * hi;
      v2f a = *(const v2f*)&Ab[kb];          // A[lm][kb], A[lm][kb+1]
      v2f bm;
      bm.x = Bb[(kb + 0) * BSTR];            // W1[kb  ][n0+lm]
      bm.y = Bb[(kb + 1) * BSTR];            // W1[kb+1][n0+lm]
      acc = __builtin_amdgcn_wmma_f32_16x16x4_f32(
          false, a, false, bm, (short)0, acc, false, false);
    }
    wait_async0();     // this wave's asyncs into the other buffer are done
    __syncthreads();   // everyone's writes visible / reads of next buf retired
  }

  // C/D layout: VGPR v, lane {hi,lm} -> (M = m0+v+8*hi, N = n0+lm).
  const int n = n0 + lm;
#pragma unroll
  for (int v = 0; v < 8; ++v) {
    const int m = m0 + v + 8 * hi;
    part[((size_t)split * B_SZ + m) * HG + n] = acc[v];
  }
}

// ---------------- Kernel 3: split-K reduce + bias + relu -------------------
__global__ void __launch_bounds__(256)
fc_reduce(const float* __restrict__ part,
          const float* __restrict__ b1,
          float* __restrict__ out) {
  const int idx = blockIdx.x * 256 + threadIdx.x;   // m * HG + n
  if (idx >= B_SZ * HG) return;
  const int m = idx / HG;
  const int n = idx % HG;
  float s = b1[n];
#pragma unroll
  for (int sp = 0; sp < KSPLIT; ++sp)
    s += part[((size_t)sp * B_SZ + m) * HG + n];
  out[(size_t)m * OUTW + n] = fmaxf(s, 0.0f);
}

// ---------------- Kernel 4: relu(l @ W2 + b2) ------------------------------
__global__ void __launch_bounds__(256)
loc_net(const float* __restrict__ l,
        const float* __restrict__ W2,
        const float* __restrict__ b2,
        float* __restrict__ out) {
  const int idx = blockIdx.x * 256 + threadIdx.x;   // b * HL + n
  if (idx >= B_SZ * HL) return;
  const int b = idx / HL;
  const int n = idx % HL;
  const float v = l[b * 2 + 0] * W2[n] + l[b * 2 + 1] * W2[HL + n] + b2[n];
  out[(size_t)b * OUTW + HG + n] = fmaxf(v, 0.0f);
}

// ---------------------------------------------------------------------------
extern "C" void kernel_launch(void* const* d_in, const int* in_sizes, int n_in,
                              void* d_out, int out_size, void* d_ws, size_t ws_size,
                              hipStream_t stream) {
  const float* x  = (const float*)d_in[0];
  const float* l  = (const float*)d_in[1];
  const float* W1 = (const float*)d_in[2];
  const float* b1 = (const float*)d_in[3];
  const float* W2 = (const float*)d_in[4];
  const float* b2 = (const float*)d_in[5];
  float* out = (float*)d_out;

  float* phi  = (float*)d_ws;                       // B_SZ * D_SZ f32 = 18.9 MB
  float* part = phi + (size_t)B_SZ * D_SZ;          // KSPLIT * 128 * 128 f32 = 512 KB

  glimpse_extract<<<B_SZ * K_SC, 256, 0, stream>>>(x, l, phi);
  glimpse_fc_wmma<<<dim3(B_SZ / 16, KSPLIT), 256, 0, stream>>>(phi, W1, part);
  fc_reduce<<<(B_SZ * HG + 255) / 256, 256, 0, stream>>>(part, b1, out);
  loc_net<<<(B_SZ * HL + 255) / 256, 256, 0, stream>>>(l, W2, b2, out);
}